// NodePart_13365938225805
// MI455X (gfx1250) — compile-verified
//
#include <hip/hip_runtime.h>
#include <hip/hip_bf16.h>
#include <math.h>

// ---------------------------------------------------------------------------
// NodePart for MI455X (gfx1250, wave32).
// phi_raw = z @ S  (S = chunk-mean selection matrix) via V_WMMA_F32_16X16X4_F32
// softmax(axis=0) -> node_weight = phi * (8 - rowsum(phi))  [since colsums==1]
// mask -> broadcast x into 8 community slabs (memory-bound: ~1.03 GB traffic).
// ---------------------------------------------------------------------------

typedef float v2f __attribute__((ext_vector_type(2)));
typedef float v4f __attribute__((ext_vector_type(4)));
typedef float v8f __attribute__((ext_vector_type(8)));

#define FEAT 256
#define NCOM 8
#define CHUNK 32
#define ROWS_PER_BLOCK 64
#define LDS_STRIDE 260   // 256 + 4-dword pad: conflict-free ds_load_b64 A reads

// ---------------- Kernel 1: phi_raw (WMMA f32), E = exp(phi_raw), partial col sums
__global__ __launch_bounds__(128) void k_phiE(const float* __restrict__ z,
                                              float* __restrict__ E,
                                              float* __restrict__ part,
                                              int N) {
    __shared__ float tile[ROWS_PER_BLOCK * LDS_STRIDE];
    __shared__ float wpart[4][NCOM];

    const int t = threadIdx.x;
    const int rowbase = blockIdx.x * ROWS_PER_BLOCK;

    // Stage 64x256 f32 tile, coalesced float4 loads (zero-fill tail rows).
    for (int j = 0; j < 32; ++j) {
        int i    = j * 128 + t;      // 0..4095 float4 slots
        int row  = i >> 6;           // 0..63
        int col4 = i & 63;           // 0..63
        v4f v = {0.f, 0.f, 0.f, 0.f};
        int gr = rowbase + row;
        if (gr < N)
            v = *reinterpret_cast<const v4f*>(z + (size_t)gr * FEAT + col4 * 4);
        float* dst = &tile[row * LDS_STRIDE + col4 * 4];
        dst[0] = v[0]; dst[1] = v[1]; dst[2] = v[2]; dst[3] = v[3];
    }
    __syncthreads();

    const int w    = t >> 5;      // wave 0..3 -> rows 16w..16w+15 of tile
    const int l    = t & 31;      // lane
    const int m    = l & 15;      // A: row / B,D: column index
    const int hi   = l >> 4;      // half-wave selects K pair (A) / M+8 (D)
    const int wrow = w * 16;

    // 64 K-steps of V_WMMA_F32_16X16X4_F32 accumulate phi_raw (cols 0..7 live).
    v8f acc = {};
    const float bval = 1.0f / (float)CHUNK;
    #pragma unroll 8
    for (int ks = 0; ks < 64; ++ks) {
        int c0 = ks >> 3;  // all 4 K's of this step live in chunk c0
        // A 16x4: VGPR0 = K {0|2}, VGPR1 = K {1|3} per half-wave (ISA 7.12.2)
        const v2f av = *reinterpret_cast<const v2f*>(
            &tile[(wrow + m) * LDS_STRIDE + ks * 4 + hi * 2]);
        v2f a; a[0] = av[0]; a[1] = av[1];
        // B 4x16: column n==c0 is 1/32, identical across K
        float bb = (m == c0) ? bval : 0.0f;
        v2f b; b[0] = bb; b[1] = bb;
        acc = __builtin_amdgcn_wmma_f32_16x16x4_f32(
            false, a, false, b, (short)0, acc, false, false);
    }

    // D layout: acc[v] = phi_raw[rowbase + wrow + v + 8*hi][col m]
    float colsum = 0.0f;
    if (m < NCOM) {
        #pragma unroll
        for (int v = 0; v < 8; ++v) {
            int grow = rowbase + wrow + v + 8 * hi;
            if (grow < N) {
                float e = expf(acc[v]);
                E[(size_t)grow * NCOM + m] = e;
                colsum += e;
            }
        }
    }
    colsum += __shfl_xor(colsum, 16, 32);      // fuse the two half-waves
    if (l < NCOM) wpart[w][l] = colsum;
    __syncthreads();
    if (t < NCOM) {                            // fixed-order: deterministic
        part[(size_t)blockIdx.x * NCOM + t] =
            wpart[0][t] + wpart[1][t] + wpart[2][t] + wpart[3][t];
    }
}

// ---------------- Kernel 2: deterministic column-sum of per-block partials
__global__ void k_reduceS(const float* __restrict__ part,
                          float* __restrict__ S, int nblk) {
    int c = threadIdx.x;
    if (c < NCOM) {
        float s = 0.0f;
        for (int b = 0; b < nblk; ++b) s += part[(size_t)b * NCOM + c];
        S[c] = s;
    }
}

// ---------------- Kernel 3: per-row mask + broadcast-scatter of x (one block/row)
__global__ __launch_bounds__(64) void k_scatter(const float* __restrict__ x,
                                                const float* __restrict__ E,
                                                const float* __restrict__ S,
                                                float* __restrict__ out,
                                                int N) {
    __shared__ unsigned sbits;
    const int n = blockIdx.x;
    const int t = threadIdx.x;

    if (t == 0) {
        float phi[NCOM], wv[NCOM];
        float r = 0.0f;
        #pragma unroll
        for (int c = 0; c < NCOM; ++c) {
            phi[c] = E[(size_t)n * NCOM + c] / S[c];
            r += phi[c];
        }
        const float f = (float)NCOM - r;      // r.sum() == C exactly
        float mx = -__builtin_inff();
        #pragma unroll
        for (int c = 0; c < NCOM; ++c) {
            wv[c] = phi[c] * f;
            mx = fmaxf(mx, wv[c]);
        }
        unsigned bits = 0;
        float* maskout = out + (size_t)NCOM * N * FEAT;
        #pragma unroll
        for (int c = 0; c < NCOM; ++c) {
            bool msk = (wv[c] == mx) || (wv[c] >= 1.0f);
            maskout[(size_t)n * NCOM + c] = msk ? 1.0f : 0.0f;
            bits |= (msk ? 1u : 0u) << c;
        }
        sbits = bits;
    }
    __syncthreads();

    const unsigned bits = sbits;
    const v4f xv = *reinterpret_cast<const v4f*>(x + (size_t)n * FEAT + t * 4);
    const v4f z4 = {0.f, 0.f, 0.f, 0.f};
    #pragma unroll
    for (int c = 0; c < NCOM; ++c) {
        v4f o = ((bits >> c) & 1u) ? xv : z4;
        // write-once 819MB stream: non-temporal so L2 keeps E/x hot instead
        __builtin_nontemporal_store(o,
            reinterpret_cast<v4f*>(out + ((size_t)c * N + n) * FEAT + t * 4));
    }
}

// ---------------------------------------------------------------------------
extern "C" void kernel_launch(void* const* d_in, const int* in_sizes, int n_in,
                              void* d_out, int out_size, void* d_ws, size_t ws_size,
                              hipStream_t stream) {
    const float* x = (const float*)d_in[0];   // [N, 256]
    const float* z = (const float*)d_in[1];   // [N, 256]
    float* out = (float*)d_out;               // [8,N,256] ++ [N,8]

    const int N = in_sizes[0] / FEAT;
    const int nblk = (N + ROWS_PER_BLOCK - 1) / ROWS_PER_BLOCK;

    float* ws   = (float*)d_ws;
    float* E    = ws;                          // N*8 floats (3.2 MB, L2-resident)
    float* part = E + (size_t)N * NCOM;        // nblk*8 floats
    float* S    = part + (size_t)nblk * NCOM;  // 8 floats

    k_phiE   <<<nblk, 128, 0, stream>>>(z, E, part, N);
    k_reduceS<<<1,    32,  0, stream>>>(part, S, nblk);
    k_scatter<<<N,    64,  0, stream>>>(x, E, S, out, N);
}